// VectorQuantizer_40269613367393
// MI455X (gfx1250) — compile-verified
//
#include <hip/hip_runtime.h>

typedef float v2f __attribute__((ext_vector_type(2)));
typedef float v8f __attribute__((ext_vector_type(8)));

#define KCODES 512
#define DDIM   64
#define CB_PITCH 65          // pad: bank = (n + k) % 64 -> conflict-free over n
#define ROWS_PER_BLOCK 128
#define HW  1024             // H*W
#define CHW 65536            // C*H*W
#define N_ZQ 16777216        // B*C*H*W
#define NBLK 2048            // 262144 rows / 128

__global__ __launch_bounds__(256) void vq_main_kernel(
    const float* __restrict__ z, const float* __restrict__ cb,
    float* __restrict__ out, float* __restrict__ ws)
{
    __shared__ float s_cb[KCODES * CB_PITCH];
    __shared__ float s_e2[KCODES];
    __shared__ int   s_idx[ROWS_PER_BLOCK];
    __shared__ float s_red[256];

    const int tid  = threadIdx.x;
    const int lane = tid & 31;
    const int wv   = tid >> 5;
    const int blk  = blockIdx.x;
    const int row0 = blk * ROWS_PER_BLOCK;
    const int b    = row0 >> 10;     // rows per image = 1024, 128 | 1024
    const int hw0  = row0 & 1023;

    // ---- stage codebook into LDS (padded), coalesced global reads ----
    for (int e = tid; e < KCODES * DDIM; e += 256) {
        int n = e >> 6, k = e & 63;
        s_cb[n * CB_PITCH + k] = cb[e];
    }
    __syncthreads();
    // ---- per-code squared norms ----
    for (int n = tid; n < KCODES; n += 256) {
        const float* r = &s_cb[n * CB_PITCH];
        float s = 0.f;
        #pragma unroll
        for (int k = 0; k < DDIM; ++k) s = fmaf(r[k], r[k], s);
        s_e2[n] = s;
    }
    __syncthreads();

    // ---- load 16-row A slab into registers (WMMA f32 16x16x4 A layout) ----
    // element (m, k) = z[b, k, hw0 + wv*16 + m]  (stride HW between k's)
    const int m    = lane & 15;
    const int half = lane >> 4;          // 0: K+{0,1}, 1: K+{2,3}
    const float* zb = z + (size_t)b * CHW + (size_t)(hw0 + wv * 16 + m);
    v2f a[16];
    #pragma unroll
    for (int k4 = 0; k4 < 16; ++k4) {
        int k = k4 * 4 + half * 2;
        a[k4].x = zb[(size_t)k * HW];
        a[k4].y = zb[(size_t)(k + 1) * HW];
    }

    float bests[8];
    int   bestidx[8];
    #pragma unroll
    for (int r = 0; r < 8; ++r) { bests[r] = 3.4e38f; bestidx[r] = 0; }

    const int ncol = lane & 15;
    for (int t = 0; t < 32; ++t) {
        const int n = t * 16 + ncol;
        const float* bcol = &s_cb[n * CB_PITCH + half * 2];
        v8f acc = {};
        #pragma unroll
        for (int k4 = 0; k4 < 16; ++k4) {
            v2f bb;
            bb.x = bcol[k4 * 4];         // adjacent pair -> ds_load_b64
            bb.y = bcol[k4 * 4 + 1];
            acc = __builtin_amdgcn_wmma_f32_16x16x4_f32(
                false, a[k4], false, bb, (short)0, acc, false, false);
        }
        const float e2n = s_e2[n];
        #pragma unroll
        for (int r = 0; r < 8; ++r) {
            float s = fmaf(-2.f, acc[r], e2n);   // ||e||^2 - 2 z.e
            if (s < bests[r]) { bests[r] = s; bestidx[r] = n; }
        }
    }

    // ---- argmin reduction across the 16 lanes of each half (rows 0-7 / 8-15) ----
    #pragma unroll
    for (int mask = 8; mask >= 1; mask >>= 1) {
        #pragma unroll
        for (int r = 0; r < 8; ++r) {
            float os = __shfl_xor(bests[r], mask, 32);
            int   oi = __shfl_xor(bestidx[r], mask, 32);
            if (os < bests[r] || (os == bests[r] && oi < bestidx[r])) {
                bests[r] = os; bestidx[r] = oi;
            }
        }
    }
    if (lane == 0 || lane == 16) {
        #pragma unroll
        for (int r = 0; r < 8; ++r) {
            int row_local = wv * 16 + half * 8 + r;
            s_idx[row_local] = bestidx[r];
            out[N_ZQ + 1 + row0 + row_local] = (float)bestidx[r];
        }
    }
    __syncthreads();

    // ---- gather z_q, coalesced store in [B,C,H,W] layout, loss partial ----
    float lsum = 0.f;
    #pragma unroll 4
    for (int it = 0; it < 32; ++it) {
        int e   = it * 256 + tid;
        int hwl = e & 127;           // consecutive tids -> consecutive hw
        int c   = e >> 7;
        int q   = s_idx[hwl];
        float v = s_cb[q * CB_PITCH + c];
        size_t g = (size_t)b * CHW + (size_t)c * HW + (size_t)(hw0 + hwl);
        out[g] = v;                  // straight-through value == z_q
        float d = v - z[g];
        lsum = fmaf(d, d, lsum);
    }
    s_red[tid] = lsum;
    __syncthreads();
    for (int s2 = 128; s2 >= 1; s2 >>= 1) {
        if (tid < s2) s_red[tid] += s_red[tid + s2];
        __syncthreads();
    }
    if (tid == 0) ws[blk] = s_red[0];
}

__global__ __launch_bounds__(256) void vq_finalize_kernel(
    const float* __restrict__ ws, float* __restrict__ out)
{
    __shared__ float s_red[256];
    int tid = threadIdx.x;
    float s = 0.f;
    for (int i = tid; i < NBLK; i += 256) s += ws[i];   // fixed order: deterministic
    s_red[tid] = s;
    __syncthreads();
    for (int s2 = 128; s2 >= 1; s2 >>= 1) {
        if (tid < s2) s_red[tid] += s_red[tid + s2];
        __syncthreads();
    }
    if (tid == 0)
        out[N_ZQ] = 1.25f * s_red[0] / (float)N_ZQ;  // codebook + beta*commitment
}

extern "C" void kernel_launch(void* const* d_in, const int* in_sizes, int n_in,
                              void* d_out, int out_size, void* d_ws, size_t ws_size,
                              hipStream_t stream)
{
    const float* z  = (const float*)d_in[0];
    const float* cb = (const float*)d_in[1];
    float* out = (float*)d_out;
    float* ws  = (float*)d_ws;     // NBLK partial sums (8 KB)

    vq_main_kernel<<<NBLK, 256, 0, stream>>>(z, cb, out, ws);
    vq_finalize_kernel<<<1, 256, 0, stream>>>(ws, out);
}